// EquivariantConvExp_37632503447670
// MI455X (gfx1250) — compile-verified
//
#include <hip/hip_runtime.h>
#include <math.h>

// ---------------------------------------------------------------------------
// EquivariantConvExp on MI455X (gfx1250, wave32, WMMA)
//
// Per flow block: z = tanh( sum_{t=0..10} K^t z / t! ), K = 7x7 "same" conv.
// Fully fused per block: one workgroup owns a 64x64 output tile, keeps the
// 124x124 (tile + 30-halo) prod region in LDS, ping-pongs 10 series terms,
// accumulates the result tile in LDS, applies tanh, writes once to HBM.
//
// Conv inner loop = exact fp32 WMMA (v_wmma_f32_16x16x4_f32): per horizontal
// tap dx, the vertical 7-tap conv of 16 output rows from 22 input rows is a
// banded-Toeplitz 16x24 (K padded) x 24x16 GEMM. The 42 Toeplitz A fragments
// (7 dx x 6 K-chunks) are patch/term-invariant, so they are precomputed once
// into LDS in WMMA per-lane v2f layout -> the hot loop is branch-free:
// ds_load_b64 (A) + ds_load_2addr_b32 (B) + v_wmma, software-pipelined.
// ---------------------------------------------------------------------------

typedef float v2f __attribute__((ext_vector_type(2)));
typedef float v8f __attribute__((ext_vector_type(8)));

#define TILE   64
#define HALO   30            // 3 px per term * 10 terms
#define RDIM   124           // TILE + 2*HALO
#define RR     (RDIM*RDIM)
#define NTERMS 10
#define IMG    256
#define NPAIR  42            // 7 dx * 6 K-chunks
// smem: wf(64) + aTab(42*32 v2f = 2688 f) + buf0 + buf1 + res
#define SMEM_FLOATS (64 + NPAIR*32*2 + 2*RR + TILE*TILE)

__global__ __launch_bounds__(256) void convexp_block_kernel(
    const float* __restrict__ in,       // [256,256,256] input z for this block
    const float* __restrict__ filters,  // [4,1,1,7,7] flat
    float* __restrict__ out,            // [256,256,256] output z
    int blk)
{
  extern __shared__ float smem[];
  float* wf   = smem;                          // 49 filter taps (+pad)
  v2f*   aT   = (v2f*)(smem + 64);             // 42*32 A fragments
  float* buf0 = smem + 64 + NPAIR * 32 * 2;    // prod ping
  float* buf1 = buf0 + RR;                     // prod pong
  float* res  = buf1 + RR;                     // 64x64 accumulated result

  const int tid  = threadIdx.x;
  const int lane = tid & 31;
  const int wave = tid >> 5;

  const int tile_id = blockIdx.x;            // 0..15 (4x4 tiles per image)
  const int img     = blockIdx.y;            // 0..255
  const int ty  = (tile_id >> 2) * TILE;     // global tile origin
  const int tx  = (tile_id &  3) * TILE;
  const int gy0 = ty - HALO;                 // region(0,0) -> global coords
  const int gx0 = tx - HALO;

  if (tid < 49) wf[tid] = filters[blk * 49 + tid];
  __syncthreads();

  // --- Precompute WMMA A fragments (Toeplitz band of the filter) ----------
  // A frag layout (16x4 f32): lane l -> M = l&15, K = {2*(l>>4), +1}.
  // A[m][k] = w[k-m][dx] for 0 <= k-m <= 6 else 0 (k = 4*ch + klocal).
  for (int idx = tid; idx < NPAIR * 32; idx += 256) {
    int pr = idx >> 5, l = idx & 31;
    int dx = pr / 6,  ch = pr % 6;
    int m  = l & 15;
    int kg = 4 * ch + 2 * (l >> 4);
    int dy0 = kg - m, dy1 = kg + 1 - m;
    v2f a;
    a.x = (dy0 >= 0 && dy0 < 7) ? wf[dy0 * 7 + dx] : 0.0f;
    a.y = (dy1 >= 0 && dy1 < 7) ? wf[dy1 * 7 + dx] : 0.0f;
    aT[idx] = a;
  }

  // --- Load input region (zero outside image = conv2d_same padding) -------
  const float* src = in + (size_t)img * IMG * IMG;
  for (int idx = tid; idx < RR; idx += 256) {
    int r = idx / RDIM, c = idx - r * RDIM;
    int gr = gy0 + r,   gc = gx0 + c;
    float v = 0.0f;
    if (gr >= 0 && gr < IMG && gc >= 0 && gc < IMG) v = src[gr * IMG + gc];
    buf0[idx] = v;
  }
  __syncthreads();

  // result starts as x (term 0 of the series).
  for (int idx = tid; idx < TILE * TILE; idx += 256) {
    int r = idx >> 6, c = idx & 63;
    res[idx] = buf0[(r + HALO) * RDIM + (c + HALO)];
  }
  __syncthreads();

  const int hi = lane >> 4;     // 0: lanes 0-15, 1: lanes 16-31
  const int nn = lane & 15;     // N col (B/C frag) / M row (A frag)

  for (int t = 1; t <= NTERMS; ++t) {
    float* bin  = (t & 1) ? buf0 : buf1;
    float* bout = (t & 1) ? buf1 : buf0;
    const int vs   = 3 * t;              // valid output region [vs, ve)
    const int ve   = RDIM - 3 * t;
    const int span = ve - vs;            // 124 - 6t
    const int np   = (span + 15) >> 4;   // 16x16 patches per dim
    const float invt = 1.0f / (float)t;

    // One wave per 16x16 output patch; waves stride over patches.
    for (int p = wave; p < np * np; p += 8) {
      const int rb = vs + (p / np) * 16;
      const int cb = vs + (p % np) * 16;
      // B per-lane base: row rb-3+2*hi, col cb+nn-3; pair (dx,ch) adds
      // dx (col) and 4*ch rows.
      const int base0 = (rb - 3 + 2 * hi) * RDIM + (cb + nn - 3);
      v8f acc = {};

      // Software-pipelined 42x (A: ds_load_b64, B: 2 x ds_load_b32, wmma).
      v2f aCur = aT[lane];
      v2f bCur;
      bCur.x = bin[base0];
      bCur.y = bin[base0 + RDIM];
      #pragma unroll
      for (int pr = 0; pr < NPAIR; ++pr) {
        v2f aNxt = {}, bNxt = {};
        if (pr < NPAIR - 1) {
          const int prn = pr + 1;
          aNxt = aT[prn * 32 + lane];
          const int adr = base0 + (prn / 6) + 4 * (prn % 6) * RDIM;
          bNxt.x = bin[adr];
          bNxt.y = bin[adr + RDIM];
        }
        acc = __builtin_amdgcn_wmma_f32_16x16x4_f32(
            false, aCur, false, bCur, (short)0, acc, false, false);
        aCur = aNxt;
        bCur = bNxt;
      }

      // Store prod_t = acc / t (zeroed outside the image = re-padding),
      // and accumulate the tile-center part into the series result.
      #pragma unroll
      for (int v = 0; v < 8; ++v) {
        int m = v + 8 * hi;                  // C/D frag row
        int r = rb + m, c = cb + nn;
        if (r < ve && c < ve) {              // r>=vs, c>=vs by construction
          int gr = gy0 + r, gc = gx0 + c;
          float val = acc[v] * invt;
          if (gr < 0 || gr >= IMG || gc < 0 || gc >= IMG) val = 0.0f;
          bout[r * RDIM + c] = val;
          int cr = r - HALO, cc = c - HALO;
          if (cr >= 0 && cr < TILE && cc >= 0 && cc < TILE)
            res[cr * TILE + cc] += val;
        }
      }
    }
    __syncthreads();
  }

  // Invertible tanh + single coalesced tile store.
  float* dst = out + (size_t)img * IMG * IMG;
  for (int idx = tid; idx < TILE * TILE; idx += 256) {
    int r = idx >> 6, c = idx & 63;
    dst[(ty + r) * IMG + (tx + c)] = tanhf(res[idx]);
  }
}

__global__ void logdet_kernel(const float* __restrict__ filters,
                              float* __restrict__ out)
{
  int b = blockIdx.x * blockDim.x + threadIdx.x;
  if (b < IMG) {
    // trace(k[:,:,3,3]) with C=1 -> k[24]; log_det -= trace * H*W per block.
    float s = filters[24] + filters[49 + 24] + filters[98 + 24] + filters[147 + 24];
    out[(size_t)IMG * IMG * IMG + b] = -65536.0f * s;
  }
}

extern "C" void kernel_launch(void* const* d_in, const int* in_sizes, int n_in,
                              void* d_out, int out_size, void* d_ws, size_t ws_size,
                              hipStream_t stream) {
  const float* x       = (const float*)d_in[0];   // [256,1,256,256] fp32
  const float* filters = (const float*)d_in[1];   // [4,1,1,7,7] fp32
  float* out = (float*)d_out;                     // z (256^3) ++ logdet (256)
  float* ws  = (float*)d_ws;                      // one 67 MB ping buffer

  dim3 grid(16, IMG);     // 4x4 tiles per image x 256 images
  dim3 block(256);        // 8 wave32
  size_t shmem = SMEM_FLOATS * sizeof(float);     // ~150 KB dynamic LDS

  // Blocks applied in reversed(range(4)) order; ping-pong ws <-> out.z
  convexp_block_kernel<<<grid, block, shmem, stream>>>(x,   filters, ws,  3);
  convexp_block_kernel<<<grid, block, shmem, stream>>>(ws,  filters, out, 2);
  convexp_block_kernel<<<grid, block, shmem, stream>>>(out, filters, ws,  1);
  convexp_block_kernel<<<grid, block, shmem, stream>>>(ws,  filters, out, 0);
  logdet_kernel<<<1, 256, 0, stream>>>(filters, out);
}